// VSSBlock3D_80101140070602
// MI455X (gfx1250) — compile-verified
//
#include <hip/hip_runtime.h>
#include <hip/hip_bf16.h>
#include <math.h>

// ---------------------------------------------------------------------------
// VSSBlock3D for MI455X (gfx1250, wave32, WMMA).
// Shapes fixed per reference: B=1, D=H=W=16 (L=4096), C=d=96, K=6, N=16, R=6.
// ---------------------------------------------------------------------------

typedef float v2f __attribute__((ext_vector_type(2)));
typedef float v8f __attribute__((ext_vector_type(8)));

#define LTOT 4096
#define CDIM 96
#define KDIR 6
#define NST  16
#define RRANK 6
#define CPROJ 38   // R + 2N
#define SEQ (CDIM * LTOT)   // per-direction plane size

static __device__ __forceinline__ v8f wmma_f32(v2f a, v2f b, v8f c) {
    // D = A(16x4 f32) x B(4x16 f32) + C(16x16 f32)
    return __builtin_amdgcn_wmma_f32_16x16x4_f32(
        /*neg_a=*/false, a, /*neg_b=*/false, b,
        /*c_mod=*/(short)0, c, /*reuse_a=*/false, /*reuse_b=*/false);
}

// --------------------------- 1) LayerNorm over C ---------------------------
__global__ void k_ln1(const float* __restrict__ x,
                      const float* __restrict__ w,
                      const float* __restrict__ b,
                      float* __restrict__ xn) {
    const int l = blockIdx.x;          // voxel
    const int lane = threadIdx.x;      // 32 lanes
    const float* xr = x + l * CDIM;
    float v0 = xr[lane], v1 = xr[lane + 32], v2 = xr[lane + 64];
    float s = v0 + v1 + v2;
    for (int m = 16; m >= 1; m >>= 1) s += __shfl_xor(s, m, 32);
    const float mean = s * (1.0f / 96.0f);
    float d0 = v0 - mean, d1 = v1 - mean, d2 = v2 - mean;
    float q = d0 * d0 + d1 * d1 + d2 * d2;
    for (int m = 16; m >= 1; m >>= 1) q += __shfl_xor(q, m, 32);
    const float rstd = rsqrtf(q * (1.0f / 96.0f) + 1e-6f);
    float* o = xn + l * CDIM;
    o[lane]      = d0 * rstd * w[lane]      + b[lane];
    o[lane + 32] = d1 * rstd * w[lane + 32] + b[lane + 32];
    o[lane + 64] = d2 * rstd * w[lane + 64] + b[lane + 64];
}

// --------------------------- 2) in_proj GEMM -------------------------------
// xz[l,e] = sum_c xn[l,c] * W[e,c]   (L=4096, E=192, C=96)
__global__ void k_inproj(const float* __restrict__ xn,
                         const float* __restrict__ W,
                         float* __restrict__ xz) {
    const int tile = blockIdx.x;           // 256 M-tiles * 12 N-tiles
    const int mt = tile & 255;
    const int nt = tile >> 8;
    const int lane = threadIdx.x;
    const int half = lane >> 4, r = lane & 15;
    const int mbase = mt * 16, nbase = nt * 16;
    v8f acc = {};
    const float* arow = xn + (mbase + r) * CDIM + 2 * half;
    const float* brow = W  + (nbase + r) * CDIM + 2 * half;
    for (int kk = 0; kk < CDIM; kk += 4) {
        v2f a = *(const v2f*)(arow + kk);
        v2f b = *(const v2f*)(brow + kk);
        acc = wmma_f32(a, b, acc);
    }
    for (int i = 0; i < 8; i++) {
        const int m = mbase + i + 8 * half;
        xz[m * 192 + nbase + r] = acc[i];
    }
}

// ------------------- 3) depthwise conv3d 3x3x3 + SiLU ----------------------
__global__ void k_conv(const float* __restrict__ xz,
                       const float* __restrict__ cw,
                       const float* __restrict__ cb,
                       float* __restrict__ xc) {
    const int idx = blockIdx.x * blockDim.x + threadIdx.x;
    if (idx >= CDIM * LTOT) return;
    const int l  = idx & (LTOT - 1);
    const int ch = idx >> 12;
    const int px = l & 15, py = (l >> 4) & 15, pz = l >> 8;
    const float* wch = cw + ch * 27;
    float acc = 0.f;
    #pragma unroll
    for (int kz = 0; kz < 3; kz++) {
        const int zz = pz + kz - 1;
        if ((unsigned)zz >= 16u) continue;
        #pragma unroll
        for (int ky = 0; ky < 3; ky++) {
            const int yy = py + ky - 1;
            if ((unsigned)yy >= 16u) continue;
            #pragma unroll
            for (int kx = 0; kx < 3; kx++) {
                const int xx = px + kx - 1;
                if ((unsigned)xx >= 16u) continue;
                acc += wch[kz * 9 + ky * 3 + kx] *
                       xz[(zz * 256 + yy * 16 + xx) * 192 + ch];
            }
        }
    }
    acc += cb[ch];
    acc = acc / (1.0f + expf(-acc));     // SiLU
    xc[ch * LTOT + l] = acc;
}

// ------------------- 4) build 6 scan orderings xs[k,ch,l] ------------------
__global__ void k_scatter(const float* __restrict__ xc, float* __restrict__ xs) {
    const int idx = blockIdx.x * blockDim.x + threadIdx.x;
    if (idx >= CDIM * LTOT) return;
    const int l0 = idx & (LTOT - 1);
    const int ch = idx >> 12;
    const int px = l0 & 15, py = (l0 >> 4) & 15, pz = l0 >> 8;
    const int l1 = (py * 16 + px) * 16 + pz;   // (H,W,D) order
    const int l2 = (px * 16 + pz) * 16 + py;   // (W,D,H) order
    const float v = xc[ch * LTOT + l0];
    float* base = xs + ch * LTOT;
    base[0 * SEQ + l0] = v;
    base[1 * SEQ + l1] = v;
    base[2 * SEQ + l2] = v;
    base[3 * SEQ + (LTOT - 1 - l0)] = v;
    base[4 * SEQ + (LTOT - 1 - l1)] = v;
    base[5 * SEQ + (LTOT - 1 - l2)] = v;
}

// --------------------------- 5) x_proj GEMM --------------------------------
// xdbl[k,c,l] = sum_d xs[k,d,l] * W[k,c,d]   (C=38 padded to 48)
__global__ void k_xproj(const float* __restrict__ xs,
                        const float* __restrict__ W,
                        float* __restrict__ xdbl) {
    int t = blockIdx.x;                 // K(6) * MT(3) * NT(256)
    const int nt = t & 255; t >>= 8;
    const int mt = t % 3;
    const int k  = t / 3;
    const int lane = threadIdx.x;
    const int half = lane >> 4, r = lane & 15;
    const int mbase = mt * 16, nbase = nt * 16;
    const float* Wk = W  + k * CPROJ * CDIM;
    const float* Xk = xs + k * SEQ;
    const int mrow = mbase + r;
    const bool mval = (mrow < CPROJ);
    v8f acc = {};
    for (int kk = 0; kk < CDIM; kk += 4) {
        const int k0 = kk + 2 * half;
        v2f a;
        if (mval) a = *(const v2f*)(Wk + mrow * CDIM + k0);
        else      { a.x = 0.f; a.y = 0.f; }
        v2f b;
        b.x = Xk[k0 * LTOT + nbase + r];
        b.y = Xk[(k0 + 1) * LTOT + nbase + r];
        acc = wmma_f32(a, b, acc);
    }
    for (int i = 0; i < 8; i++) {
        const int m = mbase + i + 8 * half;
        if (m < CPROJ) xdbl[(k * CPROJ + m) * LTOT + nbase + r] = acc[i];
    }
}

// --------------- 6) dt projection + bias + softplus (K=6 pad 8) ------------
__global__ void k_dt(const float* __restrict__ xdbl,
                     const float* __restrict__ Wd,
                     const float* __restrict__ bd,
                     float* __restrict__ dts) {
    int t = blockIdx.x;                 // K(6) * MT(6) * NT(256)
    const int nt = t & 255; t >>= 8;
    const int mt = t % 6;
    const int k  = t / 6;
    const int lane = threadIdx.x;
    const int half = lane >> 4, r = lane & 15;
    const int mbase = mt * 16, nbase = nt * 16;
    const float* Wk = Wd   + k * CDIM * RRANK;
    const float* Xk = xdbl + k * CPROJ * LTOT;
    v8f acc = {};
    #pragma unroll
    for (int kk = 0; kk < 8; kk += 4) {
        const int k0 = kk + 2 * half;
        v2f a, b;
        a.x = (k0     < RRANK) ? Wk[(mbase + r) * RRANK + k0]     : 0.f;
        a.y = (k0 + 1 < RRANK) ? Wk[(mbase + r) * RRANK + k0 + 1] : 0.f;
        b.x = (k0     < RRANK) ? Xk[k0       * LTOT + nbase + r]  : 0.f;
        b.y = (k0 + 1 < RRANK) ? Xk[(k0 + 1) * LTOT + nbase + r]  : 0.f;
        acc = wmma_f32(a, b, acc);
    }
    for (int i = 0; i < 8; i++) {
        const int m = mbase + i + 8 * half;
        float v = acc[i] + bd[k * CDIM + m];
        v = (v > 20.f) ? v : log1pf(expf(v));       // softplus
        dts[(k * CDIM + m) * LTOT + nbase + r] = v;
    }
}

// -------- 7) fused selective scan: lane n <-> state n, 2 seqs/wave ---------
__global__ void k_scan(const float* __restrict__ xs,
                       const float* __restrict__ xdbl,
                       const float* __restrict__ dts,
                       const float* __restrict__ A_log,
                       const float* __restrict__ Dsk,
                       float* __restrict__ ys) {
    const int lane = threadIdx.x;       // blockDim = 32
    const int half = lane >> 4, n = lane & 15;
    const int p = blockIdx.x * 2 + half;   // (k,ch) pair: 0..575
    const int k = p / CDIM, ch = p % CDIM;
    const int base = (k * CDIM + ch) * LTOT;
    const float* Bp = xdbl + (k * CPROJ + RRANK + n) * LTOT;
    const float* Cp = xdbl + (k * CPROJ + RRANK + NST + n) * LTOT;
    const float An    = -expf(A_log[(k * CDIM + ch) * NST + n]);
    const float dskip = Dsk[k * CDIM + ch];
    float h = 0.f;
    for (int l = 0; l < LTOT; l++) {
        const float dt = dts[base + l];
        const float xv = xs[base + l];
        h = expf(dt * An) * h + (dt * xv) * Bp[l];
        float y = h * Cp[l];
        y += __shfl_xor(y, 1, 16);
        y += __shfl_xor(y, 2, 16);
        y += __shfl_xor(y, 4, 16);
        y += __shfl_xor(y, 8, 16);
        if (n == 0) ys[base + l] = y + dskip * xv;
    }
}

// --------- 8) merge directions + inverse permutes -> voxel-major -----------
__global__ void k_merge(const float* __restrict__ ys, float* __restrict__ yt) {
    const int idx = blockIdx.x * blockDim.x + threadIdx.x;
    if (idx >= CDIM * LTOT) return;
    const int l0 = idx & (LTOT - 1);
    const int ch = idx >> 12;
    const int px = l0 & 15, py = (l0 >> 4) & 15, pz = l0 >> 8;
    const int l1 = (py * 16 + px) * 16 + pz;
    const int l2 = (px * 16 + pz) * 16 + py;
    const float* b = ys + ch * LTOT;
    float v = b[0 * SEQ + l0] + b[3 * SEQ + (LTOT - 1 - l0)]
            + b[1 * SEQ + l1] + b[4 * SEQ + (LTOT - 1 - l1)]
            + b[2 * SEQ + l2] + b[5 * SEQ + (LTOT - 1 - l2)];
    yt[l0 * CDIM + ch] = v;
}

// --------------- 9) out LayerNorm (over d) + SiLU(z) gate ------------------
__global__ void k_ln2_gate(const float* __restrict__ yt,
                           const float* __restrict__ xz,   // z at offset 96
                           const float* __restrict__ w,
                           const float* __restrict__ b,
                           float* __restrict__ g) {
    const int l = blockIdx.x;
    const int lane = threadIdx.x;
    const float* yr = yt + l * CDIM;
    float v0 = yr[lane], v1 = yr[lane + 32], v2 = yr[lane + 64];
    float s = v0 + v1 + v2;
    for (int m = 16; m >= 1; m >>= 1) s += __shfl_xor(s, m, 32);
    const float mean = s * (1.0f / 96.0f);
    float d0 = v0 - mean, d1 = v1 - mean, d2 = v2 - mean;
    float q = d0 * d0 + d1 * d1 + d2 * d2;
    for (int m = 16; m >= 1; m >>= 1) q += __shfl_xor(q, m, 32);
    const float rstd = rsqrtf(q * (1.0f / 96.0f) + 1e-5f);
    const float* zr = xz + l * 192 + CDIM;
    float* o = g + l * CDIM;
    #pragma unroll
    for (int j = 0; j < 3; j++) {
        const int c = lane + 32 * j;
        const float dv = (j == 0 ? d0 : (j == 1 ? d1 : d2));
        const float z = zr[c];
        const float silu = z / (1.0f + expf(-z));
        o[c] = (dv * rstd * w[c] + b[c]) * silu;
    }
}

// ------------------- 10) out_proj GEMM + residual add ----------------------
// out[l,c] = x[l,c] + sum_e g[l,e] * W[c,e]
__global__ void k_outproj(const float* __restrict__ g,
                          const float* __restrict__ W,
                          const float* __restrict__ x,
                          float* __restrict__ out) {
    const int tile = blockIdx.x;        // 256 M-tiles * 6 N-tiles
    const int mt = tile & 255;
    const int nt = tile >> 8;
    const int lane = threadIdx.x;
    const int half = lane >> 4, r = lane & 15;
    const int mbase = mt * 16, nbase = nt * 16;
    v8f acc = {};
    const float* arow = g + (mbase + r) * CDIM + 2 * half;
    const float* brow = W + (nbase + r) * CDIM + 2 * half;
    for (int kk = 0; kk < CDIM; kk += 4) {
        v2f a = *(const v2f*)(arow + kk);
        v2f b = *(const v2f*)(brow + kk);
        acc = wmma_f32(a, b, acc);
    }
    for (int i = 0; i < 8; i++) {
        const int m = mbase + i + 8 * half;
        const int off = m * CDIM + nbase + r;
        out[off] = x[off] + acc[i];
    }
}

// ---------------------------------------------------------------------------
extern "C" void kernel_launch(void* const* d_in, const int* in_sizes, int n_in,
                              void* d_out, int out_size, void* d_ws, size_t ws_size,
                              hipStream_t stream) {
    const float* x         = (const float*)d_in[0];
    const float* ln1_w     = (const float*)d_in[1];
    const float* ln1_b     = (const float*)d_in[2];
    const float* in_proj_w = (const float*)d_in[3];
    const float* conv_w    = (const float*)d_in[4];
    const float* conv_b    = (const float*)d_in[5];
    const float* x_proj_w  = (const float*)d_in[6];
    const float* dt_proj_w = (const float*)d_in[7];
    const float* dt_proj_b = (const float*)d_in[8];
    const float* A_log     = (const float*)d_in[9];
    const float* D_skip    = (const float*)d_in[10];
    const float* out_nw    = (const float*)d_in[11];
    const float* out_nb    = (const float*)d_in[12];
    const float* out_pw    = (const float*)d_in[13];
    float* out = (float*)d_out;

    // workspace carve-up (floats)
    float* ws   = (float*)d_ws;
    float* xn   = ws;                       // 4096*96
    float* xz   = xn   + LTOT * CDIM;       // 4096*192
    float* xc   = xz   + LTOT * 192;        // 96*4096
    float* xs   = xc   + SEQ;               // 6*96*4096
    float* xdbl = xs   + KDIR * SEQ;        // 6*38*4096
    float* dts  = xdbl + KDIR * CPROJ * LTOT; // 6*96*4096
    float* ys   = dts  + KDIR * SEQ;        // 6*96*4096
    float* yt   = ys   + KDIR * SEQ;        // 4096*96
    float* gbuf = yt   + LTOT * CDIM;       // 4096*96

    // 1) layernorm
    k_ln1<<<LTOT, 32, 0, stream>>>(x, ln1_w, ln1_b, xn);
    // 2) in_proj WMMA: 256 * 12 tiles
    k_inproj<<<256 * 12, 32, 0, stream>>>(xn, in_proj_w, xz);
    // 3) depthwise conv + SiLU
    k_conv<<<(CDIM * LTOT + 255) / 256, 256, 0, stream>>>(xz, conv_w, conv_b, xc);
    // 4) scan orderings
    k_scatter<<<(CDIM * LTOT + 255) / 256, 256, 0, stream>>>(xc, xs);
    // 5) x_proj WMMA: 6 * 3 * 256 tiles
    k_xproj<<<KDIR * 3 * 256, 32, 0, stream>>>(xs, x_proj_w, xdbl);
    // 6) dt_proj WMMA + softplus: 6 * 6 * 256 tiles
    k_dt<<<KDIR * 6 * 256, 32, 0, stream>>>(xdbl, dt_proj_w, dt_proj_b, dts);
    // 7) selective scan: 576 (k,ch) sequences, 2 per wave
    k_scan<<<288, 32, 0, stream>>>(xs, xdbl, dts, A_log, D_skip, ys);
    // 8) merge directions
    k_merge<<<(CDIM * LTOT + 255) / 256, 256, 0, stream>>>(ys, yt);
    // 9) out layernorm + gate
    k_ln2_gate<<<LTOT, 32, 0, stream>>>(yt, xz, out_nw, out_nb, gbuf);
    // 10) out_proj WMMA + residual
    k_outproj<<<256 * 6, 32, 0, stream>>>(gbuf, out_pw, x, out);
}